// DataEmbedding_19971597926793
// MI455X (gfx1250) — compile-verified
//
#include <hip/hip_runtime.h>
#include <math.h>

typedef __attribute__((ext_vector_type(2))) float v2f;
typedef __attribute__((ext_vector_type(4))) float v4f;
typedef __attribute__((ext_vector_type(8))) float v8f;

#define NNODES 307
#define NPAD   320      // 20 tiles of 16 rows
#define NEDGE  4912
#define DDIM   64
#define BATCH  8
#define TDIM   288

// ---------------------------------------------------------------------------
// WMMA fp32 GEMM:  C[NPAD x 64] = A[NPAD x K] @ B[K x 64]
// A must be zero-padded to NPAD rows -> no per-lane predication, EXEC all-1s.
// One wave per 16x16 tile; K templated so the chain of
// V_WMMA_F32_16X16X4_F32 fully unrolls (8 or 16 wmma per wave).
// A fragment (16x4 f32): lanes 0-15 hold K=k,k+1; lanes 16-31 hold K=k+2,k+3
// (one 8-byte load). C/D: VGPR i -> row i (+8 for upper lanes), col = lane&15.
// ---------------------------------------------------------------------------
template <int K>
__global__ void gemm_wmma(const float* __restrict__ A, const float* __restrict__ B,
                          float* __restrict__ C) {
  int wave = (blockIdx.x * blockDim.x + threadIdx.x) >> 5;
  int lane = threadIdx.x & 31;
  int tm = wave >> 2;                 // DDIM/16 == 4 N-tiles
  int tn = wave & 3;
  int half = lane >> 4;
  int l    = lane & 15;
  const float* Ar = A + (size_t)(tm * 16 + l) * K + half * 2;
  const float* Bc = B + tn * 16 + l + half * 2 * DDIM;
  v8f acc = {};
#pragma unroll
  for (int k = 0; k < K; k += 4) {
    v2f a = *(const v2f*)(Ar + k);    // contiguous pair -> global_load_b64
    v2f b;
    b.x = Bc[(size_t)k * DDIM];
    b.y = Bc[(size_t)k * DDIM + DDIM];
    acc = __builtin_amdgcn_wmma_f32_16x16x4_f32(false, a, false, b, (short)0, acc,
                                                false, false);
  }
#pragma unroll
  for (int i = 0; i < 8; ++i)
    C[(size_t)(tm * 16 + half * 8 + i) * DDIM + tn * 16 + l] = acc[i];
}

// node_features [307x32] -> zero-padded [320x32]
__global__ void pad_nodef(const float* __restrict__ src, float* __restrict__ dst) {
  int idx = blockIdx.x * blockDim.x + threadIdx.x;    // n*32 + c
  if (idx >= NPAD * 32) return;
  dst[idx] = ((idx >> 5) < NNODES) ? src[idx] : 0.f;
}

// nf[n,d] += b_sta[d] + ada_emb[n,d]; zero padding rows
__global__ void nf_epi(float* __restrict__ nf, const float* __restrict__ b_sta,
                       const float* __restrict__ ada) {
  int idx = blockIdx.x * blockDim.x + threadIdx.x;
  if (idx >= NPAD * DDIM) return;
  if ((idx >> 6) < NNODES) nf[idx] += b_sta[idx & 63] + ada[idx];
  else                     nf[idx] = 0.f;
}

// per (n,h): ss/st dot products; init m=-inf, den=0; zero acc row
__global__ void node_scores(const float* __restrict__ p, const float* __restrict__ asrc,
                            const float* __restrict__ atgt, float* __restrict__ ss,
                            float* __restrict__ st, float* __restrict__ m,
                            float* __restrict__ den, float* __restrict__ acc,
                            int H, int F) {
  int idx = blockIdx.x * blockDim.x + threadIdx.x;    // n*H + h
  if (idx >= NNODES * H) return;
  int n = idx / H, h = idx % H;
  const float* pr = p + (size_t)n * DDIM + h * F;
  const float* as = asrc + h * F;
  const float* at = atgt + h * F;
  float s0 = 0.f, s1 = 0.f;
  for (int f = 0; f < F; ++f) { s0 = fmaf(pr[f], as[f], s0); s1 = fmaf(pr[f], at[f], s1); }
  ss[idx] = s0;
  st[idx] = s1;
  m[idx]  = -__builtin_inff();
  den[idx] = 0.f;
  float* ar = acc + (size_t)n * DDIM + h * F;
  for (int f = 0; f < F; ++f) ar[f] = 0.f;
}

__device__ __forceinline__ void atomicMaxF(float* a, float v) {
  if (v >= 0.f) atomicMax((int*)a, __float_as_int(v));
  else          atomicMin((unsigned int*)a, __float_as_uint(v));
}

// per (e,h): e = leaky_relu(ss[src]+st[tgt]); segment max into m[tgt]
__global__ void edge_max(const int* __restrict__ ei, const float* __restrict__ ss,
                         const float* __restrict__ st, float* __restrict__ ev,
                         float* __restrict__ m, int H) {
  int idx = blockIdx.x * blockDim.x + threadIdx.x;
  if (idx >= NEDGE * H) return;
  int e = idx / H, h = idx % H;
  int s = ei[e], t = ei[NEDGE + e];
  float v = ss[s * H + h] + st[t * H + h];
  v = fmaxf(v, 0.2f * v);                             // leaky_relu(0.2)
  ev[idx] = v;
  atomicMaxF(&m[t * H + h], v);
}

// per (e,h): ex = exp(e - m[tgt]); segment sum into den[tgt]
__global__ void edge_expsum(const int* __restrict__ ei, float* __restrict__ ev,
                            const float* __restrict__ m, float* __restrict__ den, int H) {
  int idx = blockIdx.x * blockDim.x + threadIdx.x;
  if (idx >= NEDGE * H) return;
  int e = idx / H, h = idx % H;
  int t = ei[NEDGE + e];
  float ex = __expf(ev[idx] - m[t * H + h]);
  ev[idx] = ex;
  atomicAdd(&den[t * H + h], ex);
}

// per (e, c=0..63): acc[tgt, c] += p[src, c] * alpha(e, h=c/F)
__global__ void edge_scatter(const int* __restrict__ ei, const float* __restrict__ ev,
                             const float* __restrict__ den, const float* __restrict__ p,
                             float* __restrict__ acc, int H, int F) {
  int idx = blockIdx.x * blockDim.x + threadIdx.x;
  if (idx >= NEDGE * DDIM) return;
  int e = idx >> 6, c = idx & 63;
  int h = c / F;
  int s = ei[e], t = ei[NEDGE + e];
  float alpha = ev[e * H + h] / (den[t * H + h] + 1e-16f);
  atomicAdd(&acc[(size_t)t * DDIM + c], p[(size_t)s * DDIM + c] * alpha);
}

// out[n,d] = (acc + skip + bias)[n,d]  (+ elu if not last); zero padding rows
__global__ void gat_finish(const float* __restrict__ acc, const float* __restrict__ skip,
                           const float* __restrict__ bias, float* __restrict__ out,
                           int do_elu) {
  int idx = blockIdx.x * blockDim.x + threadIdx.x;    // n*64 + d
  if (idx >= NPAD * DDIM) return;
  if ((idx >> 6) >= NNODES) { out[idx] = 0.f; return; }
  float v = acc[idx] + skip[idx] + bias[idx & 63];
  if (do_elu) v = v > 0.f ? v : (__expf(v) - 1.f);
  out[idx] = v;
}

// pbt[t,d] = sinusoidal_pe(t,d) + b_val[d]   (fold bias into the PE table)
__global__ void pe_kernel(const float* __restrict__ b_val, float* __restrict__ pbt) {
  int idx = blockIdx.x * blockDim.x + threadIdx.x;    // t*32 + i
  if (idx >= TDIM * 32) return;
  int t = idx >> 5, i = idx & 31;
  float div = __expf((float)(2 * i) * (-9.210340371976184f / 64.f));  // -ln(1e4)/D
  float s, c;
  __sincosf((float)t * div, &s, &c);
  pbt[t * 64 + 2 * i]     = s + b_val[2 * i];
  pbt[t * 64 + 2 * i + 1] = c + b_val[2 * i + 1];
}

// out[b,t,n,d] = x[b,t,n]*W_val[d] + pbt[t,d] + h2[n,d]  -- 181MB NT streaming
__global__ void final_bcast(const float* __restrict__ x, const float* __restrict__ Wv,
                            const float* __restrict__ pbt, const float* __restrict__ h2,
                            float* __restrict__ out) {
  int idx = blockIdx.x * blockDim.x + threadIdx.x;    // float4 index
  const int TOT4 = BATCH * TDIM * NNODES * (DDIM / 4);
  if (idx >= TOT4) return;
  int d4  = idx & 15;
  int btn = idx >> 4;                                  // == flat x index
  int n   = btn % NNODES;
  int t   = (btn / NNODES) % TDIM;
  float xv = x[btn];
  v4f w  = ((const v4f*)Wv)[d4];
  v4f pb = ((const v4f*)pbt)[t * 16 + d4];
  v4f hh = ((const v4f*)h2)[n * 16 + d4];
  v4f o;
  o.x = fmaf(xv, w.x, pb.x + hh.x);
  o.y = fmaf(xv, w.y, pb.y + hh.y);
  o.z = fmaf(xv, w.z, pb.z + hh.z);
  o.w = fmaf(xv, w.w, pb.w + hh.w);
  __builtin_nontemporal_store(o, ((v4f*)out) + idx);
}

static inline int cdiv(int a, int b) { return (a + b - 1) / b; }

extern "C" void kernel_launch(void* const* d_in, const int* in_sizes, int n_in,
                              void* d_out, int out_size, void* d_ws, size_t ws_size,
                              hipStream_t stream) {
  const float* x     = (const float*)d_in[0];
  const float* nodef = (const float*)d_in[1];
  const int*   ei    = (const int*)d_in[2];
  // d_in[3] edge_prob unused (load_trans_prob=False)
  const float* W_val = (const float*)d_in[4];
  const float* b_val = (const float*)d_in[5];
  const float* W_sta = (const float*)d_in[6];
  const float* b_sta = (const float*)d_in[7];
  const float* ada   = (const float*)d_in[8];
  const float* gW[3]   = {(const float*)d_in[9],  (const float*)d_in[13], (const float*)d_in[17]};
  const float* gasrc[3]= {(const float*)d_in[10], (const float*)d_in[14], (const float*)d_in[18]};
  const float* gatgt[3]= {(const float*)d_in[11], (const float*)d_in[15], (const float*)d_in[19]};
  const float* gb[3]   = {(const float*)d_in[12], (const float*)d_in[16], (const float*)d_in[20]};
  const float* gskip[2]= {(const float*)d_in[21], (const float*)d_in[22]};

  float* ws   = (float*)d_ws;
  float* nf   = ws;                       // [320,64]
  float* h0   = nf   + NPAD * DDIM;
  float* h1   = h0   + NPAD * DDIM;
  float* h2   = h1   + NPAD * DDIM;
  float* p    = h2   + NPAD * DDIM;
  float* skb  = p    + NPAD * DDIM;
  float* acc  = skb  + NPAD * DDIM;
  float* ss   = acc  + NPAD * DDIM;       // [320,8]
  float* st   = ss   + NPAD * 8;
  float* m    = st   + NPAD * 8;
  float* den  = m    + NPAD * 8;
  float* ev   = den  + NPAD * 8;          // [4912,8]
  float* pbt  = ev   + NEDGE * 8;         // [288,64]
  float* nfp  = pbt  + TDIM * DDIM;       // [320,32] zero-padded node_features

  const int gemm_threads = (NPAD / 16) * (DDIM / 16) * 32;   // 80 waves = 2560
  const int gemm_blocks  = gemm_threads / 256;

  // nf = pad(node_features) @ W_sta  (+ b_sta + ada_emb); padding rows -> 0
  pad_nodef<<<cdiv(NPAD * 32, 256), 256, 0, stream>>>(nodef, nfp);
  gemm_wmma<32><<<gemm_blocks, 256, 0, stream>>>(nfp, W_sta, nf);
  nf_epi<<<cdiv(NPAD * DDIM, 256), 256, 0, stream>>>(nf, b_sta, ada);

  const int Hs[3] = {8, 8, 1};
  const int Fs[3] = {8, 8, 64};
  float* houts[3] = {h0, h1, h2};
  const float* hin = nf;
  for (int l = 0; l < 3; ++l) {
    int H = Hs[l], F = Fs[l];
    gemm_wmma<64><<<gemm_blocks, 256, 0, stream>>>(hin, gW[l], p);
    node_scores<<<cdiv(NNODES * H, 256), 256, 0, stream>>>(p, gasrc[l], gatgt[l],
                                                           ss, st, m, den, acc, H, F);
    edge_max<<<cdiv(NEDGE * H, 256), 256, 0, stream>>>(ei, ss, st, ev, m, H);
    edge_expsum<<<cdiv(NEDGE * H, 256), 256, 0, stream>>>(ei, ev, m, den, H);
    edge_scatter<<<cdiv(NEDGE * DDIM, 256), 256, 0, stream>>>(ei, ev, den, p, acc, H, F);
    const float* skipsrc;
    if (l < 2) {   // projected skip
      gemm_wmma<64><<<gemm_blocks, 256, 0, stream>>>(hin, gskip[l], skb);
      skipsrc = skb;
    } else {       // identity skip (F == 64)
      skipsrc = hin;
    }
    gat_finish<<<cdiv(NPAD * DDIM, 256), 256, 0, stream>>>(acc, skipsrc, gb[l],
                                                           houts[l], l < 2 ? 1 : 0);
    hin = houts[l];
  }

  pe_kernel<<<cdiv(TDIM * 32, 256), 256, 0, stream>>>(b_val, pbt);

  const int tot4 = BATCH * TDIM * NNODES * (DDIM / 4);
  final_bcast<<<cdiv(tot4, 256), 256, 0, stream>>>(x, W_val, pbt, h2, (float*)d_out);
}